// SnnEncoder_6957847019795
// MI455X (gfx1250) — compile-verified
//
#include <hip/hip_runtime.h>
#include <cmath>
#include <cstddef>
#include <cstdint>

typedef __attribute__((ext_vector_type(2))) float v2f;
typedef __attribute__((ext_vector_type(8))) float v8f;

#define T_BINS 32
#define HW 128
#define USE_ASYNC_STAGE 1   // global_load_async_to_lds_b128 staging path

struct Taps { float w[32]; };
struct SpikeCfg { float refk[32]; float theta; };

// ---------------------------------------------------------------------------
// PSP: causal temporal FIR along T. One lane owns one (n,c,h,w) row of 32
// contiguous floats. Triangular fully-unrolled FMA chain, taps zero-padded to
// 32 so all register indices are compile-time constants.
// ---------------------------------------------------------------------------
__global__ void psp_kernel(const float* __restrict__ in, float* __restrict__ out,
                           int rows, Taps taps) {
  int r = blockIdx.x * blockDim.x + threadIdx.x;
  if (r >= rows) return;
  const float4* ip = reinterpret_cast<const float4*>(in + (size_t)r * T_BINS);
  float x[T_BINS];
#pragma unroll
  for (int i = 0; i < 8; ++i) {
    float4 v = ip[i];
    x[4*i+0] = v.x; x[4*i+1] = v.y; x[4*i+2] = v.z; x[4*i+3] = v.w;
  }
  float y[T_BINS];
#pragma unroll
  for (int t = 0; t < T_BINS; ++t) {
    float acc = 0.f;
#pragma unroll
    for (int k = 0; k <= t; ++k)
      acc = __builtin_fmaf(taps.w[k], x[t - k], acc);
    y[t] = acc;
  }
  float4* op = reinterpret_cast<float4*>(out + (size_t)r * T_BINS);
#pragma unroll
  for (int i = 0; i < 8; ++i) {
    float4 v; v.x = y[4*i]; v.y = y[4*i+1]; v.z = y[4*i+2]; v.w = y[4*i+3];
    op[i] = v;
  }
}

// ---------------------------------------------------------------------------
// Spike generation: sequential over T per neuron row; refractory feedback as
// forward-scattered accumulation (spike at t adds refk[d] to membrane at t+d).
// ---------------------------------------------------------------------------
__global__ void spike_kernel(const float* __restrict__ p, float* __restrict__ s,
                             int rows, SpikeCfg cfg) {
  int r = blockIdx.x * blockDim.x + threadIdx.x;
  if (r >= rows) return;
  const float4* ip = reinterpret_cast<const float4*>(p + (size_t)r * T_BINS);
  float m[T_BINS];
#pragma unroll
  for (int i = 0; i < 8; ++i) {
    float4 v = ip[i];
    m[4*i+0] = v.x; m[4*i+1] = v.y; m[4*i+2] = v.z; m[4*i+3] = v.w;
  }
  float ref[T_BINS];
#pragma unroll
  for (int t = 0; t < T_BINS; ++t) ref[t] = 0.f;
  float sv[T_BINS];
#pragma unroll
  for (int t = 0; t < T_BINS; ++t) {
    float u = m[t] + ref[t];
    float sp = (u >= cfg.theta) ? 1.f : 0.f;
    sv[t] = sp;
#pragma unroll
    for (int d = 1; t + d < T_BINS; ++d)
      ref[t + d] = __builtin_fmaf(cfg.refk[d], sp, ref[t + d]);
  }
  float4* op = reinterpret_cast<float4*>(s + (size_t)r * T_BINS);
#pragma unroll
  for (int i = 0; i < 8; ++i) {
    float4 v; v.x = sv[4*i]; v.y = sv[4*i+1]; v.z = sv[4*i+2]; v.w = sv[4*i+3];
    op[i] = v;
  }
}

// ---------------------------------------------------------------------------
// 3x3 conv (pad 1), Co = 16, shared across time, WMMA f32 16x16x4.
//
// Block = 128 threads = 4 waves = 4 consecutive pixels (same image row).
// LDS tile: all CT channels x 3 rows x 7 cols (4 pixels + halo + 1 pad col so
// the kw=3 padded B row reads staged finite data) x 32 T floats -> coalesced
// cooperative staging, boundary zeros written once.
// K dimension re-chunked as (ci,kh) with kw padded 3->4 (pad weight = 0), so
// the inner loop is branchless: every operand is a constant-offset LDS read.
// ---------------------------------------------------------------------------
__global__ void conv3x3_wmma_kernel(const float* __restrict__ in0, int C0,
                                    const float* __restrict__ in1, int C1,
                                    const float* __restrict__ wgt,
                                    float* __restrict__ out,
                                    int CT, int totalPix) {
  __shared__ float tile[18 * 21 * 32];   // [ci][dy(3)][x(7)][t]  47.25 KB
  __shared__ float wlds[16 * 216];       // [co][(ci*3+kh)*4 + kw] 13.5 KB

  int pixBase = blockIdx.x << 2;
  int n     = pixBase >> 14;             // / (128*128)
  int rem   = pixBase & 16383;
  int h     = rem >> 7;
  int wbase = rem & 127;                 // multiple of 4, never straddles a row

  // ---- stage weights (repacked, kw padded to 4 with zeros) ----
  const int KP = CT * 12;
  for (int co = 0; co < 16; ++co)
    for (int k = threadIdx.x; k < KP; k += blockDim.x) {
      int chunk = k >> 2, kw = k & 3;
      int ci = chunk / 3;
      int kh = chunk - ci * 3;
      wlds[co * KP + k] = (kw < 3) ? wgt[(co * CT + ci) * 9 + kh * 3 + kw] : 0.f;
    }

  // ---- stage input tile (coalesced float4 rows, zeros at boundaries) ----
  const int ROWS = CT * 21;              // CT * 3dy * 7x
  for (int i = threadIdx.x; i < ROWS * 8; i += blockDim.x) {
    int row = i >> 3;
    int q   = i & 7;
    int ci  = row / 21;
    int r2  = row - ci * 21;
    int dy  = r2 / 7;
    int x   = r2 - dy * 7;
    int y   = h + dy - 1;
    int xg  = wbase + x - 1;
    bool inb = ((unsigned)y < (unsigned)HW) && ((unsigned)xg < (unsigned)HW);
    const float* base = (ci < C0)
        ? in0 + ((((size_t)n * C0 + ci) * HW + y) * HW + xg) * T_BINS
        : in1 + ((((size_t)n * C1 + (ci - C0)) * HW + y) * HW + xg) * T_BINS;
#if USE_ASYNC_STAGE
    if (inb) {
      uint32_t la = (uint32_t)(uintptr_t)&tile[row * T_BINS + q * 4];
      unsigned long long ga = (unsigned long long)(uintptr_t)(base + q * 4);
      asm volatile("global_load_async_to_lds_b128 %0, %1, off"
                   :: "v"(la), "v"(ga) : "memory");
    } else {
      float4 z = {0.f, 0.f, 0.f, 0.f};
      *reinterpret_cast<float4*>(&tile[row * T_BINS + q * 4]) = z;
    }
#else
    float4 v = {0.f, 0.f, 0.f, 0.f};
    if (inb) v = *reinterpret_cast<const float4*>(base + q * 4);
    *reinterpret_cast<float4*>(&tile[row * T_BINS + q * 4]) = v;
#endif
  }
#if USE_ASYNC_STAGE
  asm volatile("s_wait_asynccnt 0x0" ::: "memory");
#endif
  __syncthreads();

  // ---- per-wave GEMM: D(16co x 32t) = W(16 x K) * im2col(K x 32) ----
  int wave = pixBase + (threadIdx.x >> 5);
  if (wave < totalPix) {
    int lane  = threadIdx.x & 31;
    int halfl = lane & 31 & 15;
    bool hi   = lane >= 16;
    int hi2   = hi ? 2 : 0;
    int px    = threadIdx.x >> 5;        // pixel offset in block; x=w-1 -> col px

    v8f acc0 = {0,0,0,0,0,0,0,0};
    v8f acc1 = {0,0,0,0,0,0,0,0};

    for (int ci = 0; ci < CT; ++ci) {
#pragma unroll
      for (int kh = 0; kh < 3; ++kh) {
        int kb = ((ci * 3 + kh) << 2) + hi2;
        v2f a;
        a.x = wlds[halfl * KP + kb];
        a.y = wlds[halfl * KP + kb + 1];
        const float* trow = &tile[(ci * 21 + kh * 7 + px) * T_BINS];
        int xo = hi2 << 5;               // lanes>=16 start at col px+2 (kw=2)
        v2f b0, b1;
        b0.x = trow[xo + halfl];         // kw = hi?2:0
        b0.y = trow[xo + 32 + halfl];    // kw = hi?3(pad, x0 weight):1
        b1.x = trow[xo + 16 + halfl];
        b1.y = trow[xo + 48 + halfl];
        acc0 = __builtin_amdgcn_wmma_f32_16x16x4_f32(false, a, false, b0, (short)0, acc0, false, false);
        acc1 = __builtin_amdgcn_wmma_f32_16x16x4_f32(false, a, false, b1, (short)0, acc1, false, false);
      }
    }

    // C/D layout: VGPR v, lanes 0-15 -> M=v, N=lane; lanes 16-31 -> M=v+8.
    int cobase = hi ? 8 : 0;
    const size_t chStride = (size_t)HW * HW * T_BINS;
    size_t pixofs = (((size_t)h) * HW + (wbase + px)) * T_BINS;
    size_t nbase  = (size_t)n * 16 * chStride;
#pragma unroll
    for (int v = 0; v < 8; ++v) {
      size_t o = nbase + (size_t)(cobase + v) * chStride + pixofs + halfl;
      out[o]      = acc0[v];
      out[o + 16] = acc1[v];
    }
  }
}

// ---------------------------------------------------------------------------
// 1x1 conv, Co = 32 (2 M-tiles), K = 18 (16ch pspS2 ++ 2ch psp0), WMMA f32.
// Branchless fetch: k clamped to 17 (rows 18/19 carry zero weights, so the
// clamped duplicate B value contributes nothing). Streams the 256 MB output
// with nontemporal stores to spare the 192 MB L2.
// ---------------------------------------------------------------------------
__device__ __forceinline__ float fetch1x1(const float* __restrict__ in0,
                                          const float* __restrict__ in1,
                                          int n, size_t pix, int kk, int t) {
  int kc = kk > 17 ? 17 : kk;
  const size_t chS = (size_t)HW * HW * T_BINS;
  const float* p = (kc < 16) ? in0 + ((size_t)n * 16 + kc) * chS
                             : in1 + ((size_t)n * 2 + (kc - 16)) * chS;
  return p[pix + t];
}

__global__ void conv1x1_wmma_kernel(const float* __restrict__ in0,
                                    const float* __restrict__ in1,
                                    const float* __restrict__ wgt,   // (32,18)
                                    float* __restrict__ out,
                                    int totalPix) {
  __shared__ float wlds[32 * 20];
  for (int co = 0; co < 32; ++co)
    for (int k = threadIdx.x; k < 20; k += blockDim.x)
      wlds[co * 20 + k] = (k < 18) ? wgt[co * 18 + k] : 0.f;
  __syncthreads();

  int wave = blockIdx.x * (blockDim.x >> 5) + (threadIdx.x >> 5);
  if (wave >= totalPix) return;
  int lane  = threadIdx.x & 31;
  int halfl = lane & 15;
  bool hi   = lane >= 16;

  int n   = wave >> 14;
  int rem = wave & 16383;
  int h   = rem >> 7;
  int w   = rem & 127;
  size_t pix = (((size_t)h) * HW + w) * T_BINS;

  v8f z = {0,0,0,0,0,0,0,0};
  v8f acc00 = z, acc01 = z, acc10 = z, acc11 = z;

#pragma unroll
  for (int c = 0; c < 5; ++c) {
    int kb = (c << 2) + (hi ? 2 : 0);
    v2f b0, b1;
    b0.x = fetch1x1(in0, in1, n, pix, kb,     halfl);
    b0.y = fetch1x1(in0, in1, n, pix, kb + 1, halfl);
    b1.x = fetch1x1(in0, in1, n, pix, kb,     halfl + 16);
    b1.y = fetch1x1(in0, in1, n, pix, kb + 1, halfl + 16);
    v2f a0, a1;
    a0.x = wlds[halfl * 20 + kb];
    a0.y = wlds[halfl * 20 + kb + 1];
    a1.x = wlds[(16 + halfl) * 20 + kb];
    a1.y = wlds[(16 + halfl) * 20 + kb + 1];
    acc00 = __builtin_amdgcn_wmma_f32_16x16x4_f32(false, a0, false, b0, (short)0, acc00, false, false);
    acc01 = __builtin_amdgcn_wmma_f32_16x16x4_f32(false, a0, false, b1, (short)0, acc01, false, false);
    acc10 = __builtin_amdgcn_wmma_f32_16x16x4_f32(false, a1, false, b0, (short)0, acc10, false, false);
    acc11 = __builtin_amdgcn_wmma_f32_16x16x4_f32(false, a1, false, b1, (short)0, acc11, false, false);
  }

  int cobase = hi ? 8 : 0;
  const size_t chS = (size_t)HW * HW * T_BINS;
  size_t nbase = (size_t)n * 32 * chS;
#pragma unroll
  for (int v = 0; v < 8; ++v) {
    size_t o0 = nbase + (size_t)(cobase + v) * chS + pix + halfl;
    size_t o1 = nbase + (size_t)(16 + cobase + v) * chS + pix + halfl;
    __builtin_nontemporal_store(acc00[v], &out[o0]);
    __builtin_nontemporal_store(acc01[v], &out[o0 + 16]);
    __builtin_nontemporal_store(acc10[v], &out[o1]);
    __builtin_nontemporal_store(acc11[v], &out[o1 + 16]);
  }
}

// ---------------------------------------------------------------------------
// Host side
// ---------------------------------------------------------------------------
static void alphaTapsHost(double tau, double mult, float* w) {
  for (int t = 0; t < 32; ++t) w[t] = 0.f;
  for (int t = 0; t < 32; ++t) {
    double v = mult * ((double)t / tau) * std::exp(1.0 - (double)t / tau);
    if (std::fabs(v) < 1e-6 && (double)t > tau) break;
    w[t] = (float)v;
  }
}

extern "C" void kernel_launch(void* const* d_in, const int* in_sizes, int n_in,
                              void* d_out, int out_size, void* d_ws, size_t ws_size,
                              hipStream_t stream) {
  const float* spikeInput = (const float*)d_in[0]; // (4,2,128,128,32)
  const float* w1 = (const float*)d_in[1];         // (16,2,3,3)
  const float* w2 = (const float*)d_in[2];         // (16,18,3,3)
  const float* w3 = (const float*)d_in[3];         // (32,18,1,1)
  float* out = (float*)d_out;                      // (4,32,128,128,32)

  const size_t PSP0_BYTES = (size_t)4 * 2  * HW * HW * T_BINS * sizeof(float); // 16 MiB
  const size_t BUF_BYTES  = (size_t)4 * 16 * HW * HW * T_BINS * sizeof(float); // 128 MiB
  if (ws_size < PSP0_BYTES + 2 * BUF_BYTES) return;
  float* psp0 = (float*)d_ws;
  float* bufA = (float*)((char*)d_ws + PSP0_BYTES);
  float* bufB = (float*)((char*)d_ws + PSP0_BYTES + BUF_BYTES);

  Taps srm0, srm1, srm2;
  alphaTapsHost(1.0, 1.0, srm0.w);
  alphaTapsHost(2.0, 1.0, srm1.w);
  alphaTapsHost(4.0, 1.0, srm2.w);
  SpikeCfg sc0, sc1;
  sc0.theta = 3.f; alphaTapsHost(1.0, -3.0, sc0.refk);   // -scaleRef*theta, tauRef=1
  sc1.theta = 5.f; alphaTapsHost(2.0, -5.0, sc1.refk);   // tauRef=2

  const int rows_in  = 4 * 2  * HW * HW;  // 131072
  const int rows_mid = 4 * 16 * HW * HW;  // 1048576
  const int pix      = 4 * HW * HW;       // 65536

  // psp0 = psp(spikeInput, SRM0)
  psp_kernel<<<rows_in / 256, 256, 0, stream>>>(spikeInput, psp0, rows_in, srm0);
  // p1 = conv3x3(psp0, w1) -> bufA       (CT = 2)
  conv3x3_wmma_kernel<<<pix / 4, 128, 0, stream>>>(psp0, 2, nullptr, 0, w1, bufA, 2, pix);
  // s1 = spike0(p1) -> bufB
  spike_kernel<<<rows_mid / 256, 256, 0, stream>>>(bufA, bufB, rows_mid, sc0);
  // pspS1 = psp(s1, SRM1) -> bufA
  psp_kernel<<<rows_mid / 256, 256, 0, stream>>>(bufB, bufA, rows_mid, srm1);
  // p2 = conv3x3(concat(pspS1, psp0), w2) -> bufB   (CT = 18)
  conv3x3_wmma_kernel<<<pix / 4, 128, 0, stream>>>(bufA, 16, psp0, 2, w2, bufB, 18, pix);
  // s2 = spike1(p2) -> bufA
  spike_kernel<<<rows_mid / 256, 256, 0, stream>>>(bufB, bufA, rows_mid, sc1);
  // pspS2 = psp(s2, SRM2) -> bufB
  psp_kernel<<<rows_mid / 256, 256, 0, stream>>>(bufA, bufB, rows_mid, srm2);
  // out = conv1x1(concat(pspS2, psp0), w3)
  conv1x1_wmma_kernel<<<pix / 8, 256, 0, stream>>>(bufB, psp0, w3, out, pix);
}